// GDLoss_v1_76630806495957
// MI455X (gfx1250) — compile-verified
//
#include <hip/hip_runtime.h>
#include <hip/hip_bf16.h>

#ifndef GD_EPS
#define GD_EPS 1e-6f
#endif

// Per-box KL-based Gaussian distance loss (TAU = 1, LOSS_WEIGHT = 1).
__device__ __forceinline__ float gd_box_loss(
    float px, float py, float pw, float ph, float pr,
    float tx, float ty, float tw, float th, float tr)
{
    // clip w/h to [1e-7, 1e7]
    pw = fminf(fmaxf(pw, 1e-7f), 1e7f);
    ph = fminf(fmaxf(ph, 1e-7f), 1e7f);
    tw = fminf(fmaxf(tw, 1e-7f), 1e7f);
    th = fminf(fmaxf(th, 1e-7f), 1e7f);

    // hardware transcendentals: v_cos_f32 / v_sin_f32 (input range ~[-pi,pi])
    float cp = __cosf(pr), sp = __sinf(pr);
    float ct = __cosf(tr), st = __sinf(tr);

    float ap = 0.25f * pw * pw, bp = 0.25f * ph * ph;
    float at = 0.25f * tw * tw, bt = 0.25f * th * th;

    float p11 = ap * cp * cp + bp * sp * sp;
    float p12 = (ap - bp) * sp * cp;
    float p22 = ap * sp * sp + bp * cp * cp;

    float t11 = at * ct * ct + bt * st * st;
    float t12 = (at - bt) * st * ct;
    float t22 = at * st * st + bt * ct * ct;

    float det_p = p11 * p22 - p12 * p12;
    float det_t = t11 * t22 - t12 * t12;

    float dx = px - tx;
    float dy = py - ty;

    float inv_det_t = __builtin_amdgcn_rcpf(det_t);              // v_rcp_f32
    float term1 = (t22 * dx * dx - 2.0f * t12 * dx * dy + t11 * dy * dy) * inv_det_t;
    float trace = (t22 * p11 - 2.0f * t12 * p12 + t11 * p22) * inv_det_t;
    float lg    = __logf(det_t) - __logf(det_p);                 // 2x v_log_f32

    float dis = term1 + trace + lg - 2.0f;
    float kl  = fmaxf(dis, GD_EPS);
    float sq  = __builtin_amdgcn_sqrtf(kl);                      // v_sqrt_f32
    return 1.0f - __builtin_amdgcn_rcpf(1.0f + sq);              // v_rcp_f32
}

// Each thread handles 4 consecutive boxes (80 bytes = 5 x float4, always
// 16B aligned) so all global traffic is b128 loads/stores.
__global__ void __launch_bounds__(256)
gd_loss_pack4_kernel(const float* __restrict__ pred,
                     const float* __restrict__ target,
                     float* __restrict__ out,
                     int n)
{
    const int t     = blockIdx.x * blockDim.x + threadIdx.x;
    const int packs = n >> 2;

    if (t < packs) {
        const size_t base = (size_t)t * 20;   // 4 boxes * 5 floats

        // Prefetch one block-stride ahead (256 threads * 80B = 20480B).
        // Speculative prefetch is OOB-safe on gfx1250 (global_prefetch_b8).
        __builtin_prefetch(pred   + base + 5120, 0, 1);
        __builtin_prefetch(target + base + 5120, 0, 1);

        const float4* pp = reinterpret_cast<const float4*>(pred + base);
        const float4* tp = reinterpret_cast<const float4*>(target + base);

        const float4 p0 = pp[0], p1 = pp[1], p2 = pp[2], p3 = pp[3], p4 = pp[4];
        const float4 q0 = tp[0], q1 = tp[1], q2 = tp[2], q3 = tp[3], q4 = tp[4];

        float4 o;
        // box 0: floats 0..4
        o.x = gd_box_loss(p0.x, p0.y, p0.z, p0.w, p1.x,
                          q0.x, q0.y, q0.z, q0.w, q1.x);
        // box 1: floats 5..9
        o.y = gd_box_loss(p1.y, p1.z, p1.w, p2.x, p2.y,
                          q1.y, q1.z, q1.w, q2.x, q2.y);
        // box 2: floats 10..14
        o.z = gd_box_loss(p2.z, p2.w, p3.x, p3.y, p3.z,
                          q2.z, q2.w, q3.x, q3.y, q3.z);
        // box 3: floats 15..19
        o.w = gd_box_loss(p3.w, p4.x, p4.y, p4.z, p4.w,
                          q3.w, q4.x, q4.y, q4.z, q4.w);

        reinterpret_cast<float4*>(out)[t] = o;   // global_store_b128 (16B aligned)
    }

    // Scalar tail for n not divisible by 4 (N = 4,000,000 -> empty).
    if (t == 0) {
        for (int i = packs << 2; i < n; ++i) {
            const float* p = pred   + (size_t)i * 5;
            const float* q = target + (size_t)i * 5;
            out[i] = gd_box_loss(p[0], p[1], p[2], p[3], p[4],
                                 q[0], q[1], q[2], q[3], q[4]);
        }
    }
}

extern "C" void kernel_launch(void* const* d_in, const int* in_sizes, int n_in,
                              void* d_out, int out_size, void* d_ws, size_t ws_size,
                              hipStream_t stream) {
    const float* pred   = (const float*)d_in[0];   // [N,5] f32
    const float* target = (const float*)d_in[1];   // [N,5] f32
    // d_in[2] = weight [N] f32 — unused by the reference computation.
    float* out = (float*)d_out;                    // [N,1] f32

    const int n = in_sizes[2];                     // weight's flat size == N
    if (n <= 0) return;

    const int threads = (n + 3) >> 2;              // one thread per 4-box pack
    const int block   = 256;                       // 8 wave32 waves per block
    const int grid    = (threads + block - 1) / block;

    gd_loss_pack4_kernel<<<grid, block, 0, stream>>>(pred, target, out, n);
}